// ResidualBlock_25348896981728
// MI455X (gfx1250) — compile-verified
//
#include <hip/hip_runtime.h>
#include <hip/hip_bf16.h>

// ---------------------------------------------------------------------------
// Mamba residual block, MI455X (gfx1250, wave32, WMMA + TDM async tile loads).
// fp32 everywhere; GEMMs use V_WMMA_F32_16X16X4_F32 with double-buffered LDS
// tiles fed by TENSOR_LOAD_TO_LDS (Tensor Data Mover, TENSORcnt).
// ---------------------------------------------------------------------------

#define DM    2048   // d_model
#define DI    4096   // d_inner
#define DS    16     // d_state
#define DTR   128    // dt_rank
#define SEQ   2048   // seq_len
#define NPAD  256    // x_proj columns padded 160 -> 256
#define LNEPS 1e-5f

typedef float v2f __attribute__((ext_vector_type(2)));
typedef float v8f __attribute__((ext_vector_type(8)));
typedef unsigned int v4u __attribute__((ext_vector_type(4)));
typedef int v8i __attribute__((ext_vector_type(8)));
typedef int v4i __attribute__((ext_vector_type(4)));

// ---------------- workspace layout (float offsets) -------------------------
#define WS_H     ((size_t)0)                       // 2048x2048
#define WS_XR    (WS_H    + (size_t)SEQ * DM)      // 2048x8192 (xm | r); cols 0..4095 reused for ys
#define WS_XM    (WS_XR   + (size_t)SEQ * 2 * DI)  // 2048x4096 conv+silu output
#define WS_DELTA (WS_XM   + (size_t)SEQ * DI)      // 2048x4096 delta; later reused as gated y
#define WS_XDBL  (WS_DELTA+ (size_t)SEQ * DI)      // 2048x256  padded x_dbl
#define WS_XPROJ (WS_XDBL + (size_t)SEQ * NPAD)    // 4096x256  padded x_proj_w

// ======================= TDM 2D tile load helper ===========================
// Builds a Tensor DMA Descriptor (D#) per CDNA5 ISA and
// issues TENSOR_LOAD_TO_LDS. 4-byte elements, 2D tile, optional LDS padding
// (pad_interval/pad_amount codes per D# group1 encoding).
__device__ __forceinline__ void tdm_load_2d_f32(
    const float* gsrc, unsigned int lds_byte_off,
    unsigned int tile_w, unsigned int tile_h,          // tile dims (elements)
    unsigned int tdim0, unsigned int tdim1,            // tensor extents for OOB
    unsigned long long row_stride_elems,
    int pad_en, unsigned int pad_interval_code, unsigned int pad_amount_code)
{
    unsigned long long ga = (unsigned long long)(uintptr_t)gsrc;

    v4u g0;
    g0[0] = 1u;                                        // count=1 (valid), user mode
    g0[1] = lds_byte_off;                              // lds_addr
    g0[2] = (unsigned int)ga;                          // global_addr[31:0]
    g0[3] = (unsigned int)((ga >> 32) & 0x01FFFFFFu)   // global_addr[56:32]
          | (2u << 30);                                // type = 2 ("image")

    unsigned int w0 = (2u << 16);                      // data_size = 2 -> 4 bytes
    if (pad_en)
        w0 |= (1u << 20)                               // pad_enable
            | (pad_interval_code << 22)                // [24:22]
            | (pad_amount_code << 25);                 // [31:25]

    v8i g1;
    g1[0] = (int)w0;                                               // mask=0, flags
    g1[1] = (int)((tdim0 & 0xFFFFu) << 16);                        // tensor_dim0[15:0]
    g1[2] = (int)((tdim0 >> 16) | ((tdim1 & 0xFFFFu) << 16));      // td0 hi | td1 lo
    g1[3] = (int)((tdim1 >> 16) | ((tile_w & 0xFFFFu) << 16));     // td1 hi | tile_dim0
    g1[4] = (int)(tile_h & 0xFFFFu);                               // tile_dim1 (tile_dim2=0)
    g1[5] = (int)(unsigned int)row_stride_elems;                   // dim0_stride lo32
    g1[6] = (int)(unsigned int)((row_stride_elems >> 32) & 0xFFFFu); // hi16 | dim1_stride lo16=0
    g1[7] = 0;                                                     // dim1_stride hi32

    v4i g2 = {0, 0, 0, 0};
    v4i g3 = {0, 0, 0, 0};
#if __clang_major__ >= 23
    v8i g1b = {0, 0, 0, 0, 0, 0, 0, 0};
    __builtin_amdgcn_tensor_load_to_lds(g0, g1, g2, g3, g1b, 0);
#else
    __builtin_amdgcn_tensor_load_to_lds(g0, g1, g2, g3, 0);
#endif
}

// ======================= residual + layernorm ==============================
__global__ __launch_bounds__(256) void k_res_ln(
    const float* __restrict__ x, const float* __restrict__ resid,
    const float* __restrict__ scale, const float* __restrict__ bias,
    float* __restrict__ res_out, float* __restrict__ h_out)
{
    const int row = blockIdx.x;
    const int tid = threadIdx.x;
    __shared__ float sa[8], sb[8];

    float vals[DM / 256];
    float s = 0.f, sq = 0.f;
#pragma unroll
    for (int i = 0; i < DM / 256; ++i) {
        int c = tid + i * 256;
        float v = x[(size_t)row * DM + c] + resid[(size_t)row * DM + c];
        vals[i] = v;
        res_out[(size_t)row * DM + c] = v;
        s += v; sq += v * v;
    }
    for (int off = 16; off; off >>= 1) {
        s  += __shfl_xor(s,  off, 32);
        sq += __shfl_xor(sq, off, 32);
    }
    int lane = tid & 31, w = tid >> 5;
    if (lane == 0) { sa[w] = s; sb[w] = sq; }
    __syncthreads();
    if (w == 0) {
        s  = (lane < 8) ? sa[lane] : 0.f;
        sq = (lane < 8) ? sb[lane] : 0.f;
        for (int off = 4; off; off >>= 1) {
            s  += __shfl_xor(s,  off, 32);
            sq += __shfl_xor(sq, off, 32);
        }
        if (lane == 0) { sa[0] = s; sb[0] = sq; }
    }
    __syncthreads();
    float mu  = sa[0] * (1.f / DM);
    float var = sb[0] * (1.f / DM) - mu * mu;
    float rs  = rsqrtf(var + LNEPS);
#pragma unroll
    for (int i = 0; i < DM / 256; ++i) {
        int c = tid + i * 256;
        h_out[(size_t)row * DM + c] = (vals[i] - mu) * rs * scale[c] + bias[c];
    }
}

// ======================= fp32 WMMA GEMM (TDM double-buffered) ==============
// C[M,N] = A[M,K] * B[K,N]; M,N multiples of 128, K multiple of 32.
// Block: 256 threads = 8 waves; block tile 128x128; wave tile 64x32 (4x2 WMMA).
// Wave 0 drives the TDM: async-loads tile kt+1 while all waves compute on kt.
#define BM 128
#define BN 128
#define BK 32
#define ASTRIDE (BK + 1)          // TDM pad: +1 dword per 32-dword row
#define ALDS (BM * ASTRIDE)       // 4224 floats

__global__ __launch_bounds__(256) void k_wmma_gemm_f32(
    const float* __restrict__ A, const float* __restrict__ B,
    float* __restrict__ C, int M, int N, int K, int lda, int ldb, int ldc)
{
    __shared__ float As[2][ALDS];     // row-major, stride 33 (conflict-free)
    __shared__ float Bs[2][BK * BN];  // row-major

    const int tid  = threadIdx.x;
    const int lane = tid & 31;
    const int wave = tid >> 5;
    const int lm   = lane & 15;
    const int half = lane >> 4;         // 0: k0/k1 ; 1: k2/k3
    const int wm   = (wave & 1) * 64;
    const int wn   = (wave >> 1) * 32;
    const int row0 = blockIdx.y * BM;
    const int col0 = blockIdx.x * BN;

    const v8f vzero = {0.f,0.f,0.f,0.f,0.f,0.f,0.f,0.f};
    v8f acc[4][2];
#pragma unroll
    for (int i = 0; i < 4; ++i)
#pragma unroll
        for (int j = 0; j < 2; ++j) acc[i][j] = vzero;

    const int nkt = K / BK;

    // Prologue: DMA tile 0 into buffer 0 (A padded to stride 33, B dense).
    if (wave == 0) {
        tdm_load_2d_f32(A + (size_t)row0 * lda,
                        (unsigned int)(size_t)&As[0][0],
                        BK, BM, (unsigned int)K, (unsigned int)(M - row0),
                        (unsigned long long)lda,
                        1, 4 /*interval=32 dwords*/, 0 /*pad=1 dword*/);
        tdm_load_2d_f32(B + col0,
                        (unsigned int)(size_t)&Bs[0][0],
                        BN, BK, (unsigned int)(N - col0), (unsigned int)K,
                        (unsigned long long)ldb, 0, 0, 0);
        __builtin_amdgcn_s_wait_tensorcnt(0);
    }
    __syncthreads();

    for (int kt = 0; kt < nkt; ++kt) {
        const int cur = kt & 1;
        const int nxt = cur ^ 1;
        const int k0n = (kt + 1) * BK;

        // Kick off next tile's DMA while this tile is consumed.
        if (wave == 0 && kt + 1 < nkt) {
            tdm_load_2d_f32(A + (size_t)row0 * lda + k0n,
                            (unsigned int)(size_t)&As[nxt][0],
                            BK, BM, (unsigned int)(K - k0n), (unsigned int)(M - row0),
                            (unsigned long long)lda, 1, 4, 0);
            tdm_load_2d_f32(B + (size_t)k0n * ldb + col0,
                            (unsigned int)(size_t)&Bs[nxt][0],
                            BN, BK, (unsigned int)(N - col0), (unsigned int)(K - k0n),
                            (unsigned long long)ldb, 0, 0, 0);
        }

        const float* __restrict__ Ac = As[cur];
        const float* __restrict__ Bc = Bs[cur];
#pragma unroll
        for (int kk = 0; kk < 8; ++kk) {
            const int kb = kk * 4 + half * 2;   // this half-wave's K pair
            v2f a[4], b[2];
#pragma unroll
            for (int i = 0; i < 4; ++i) {
                const int m = wm + i * 16 + lm;
                a[i].x = Ac[m * ASTRIDE + kb];
                a[i].y = Ac[m * ASTRIDE + kb + 1];
            }
#pragma unroll
            for (int j = 0; j < 2; ++j) {
                const int n = wn + j * 16 + lm;
                b[j].x = Bc[(kb + 0) * BN + n];
                b[j].y = Bc[(kb + 1) * BN + n];
            }
#pragma unroll
            for (int i = 0; i < 4; ++i)
#pragma unroll
                for (int j = 0; j < 2; ++j)
                    acc[i][j] = __builtin_amdgcn_wmma_f32_16x16x4_f32(
                        false, a[i], false, b[j], (short)0, acc[i][j], false, false);
        }

        if (wave == 0 && kt + 1 < nkt)
            __builtin_amdgcn_s_wait_tensorcnt(0);
        __syncthreads();
    }

    // ---- store C (16x16 f32 layout: VGPR v -> row v + 8*half) -------------
#pragma unroll
    for (int i = 0; i < 4; ++i)
#pragma unroll
        for (int j = 0; j < 2; ++j) {
            const int cm = row0 + wm + i * 16 + half * 8;
            const int cn = col0 + wn + j * 16 + lm;
#pragma unroll
            for (int v = 0; v < 8; ++v)
                C[(size_t)(cm + v) * ldc + cn] = acc[i][j][v];
        }
}

// ======================= depthwise causal conv + SiLU ======================
__global__ __launch_bounds__(256) void k_conv_silu(
    const float* __restrict__ xr,      // 2048 x 8192, xm in cols [0,4096)
    const float* __restrict__ conv_w,  // 4 x 1 x 4096
    const float* __restrict__ conv_b,  // 4096
    float* __restrict__ xm_out)        // 2048 x 4096
{
    const size_t i = (size_t)blockIdx.x * 256 + threadIdx.x;  // over SEQ*DI
    const int l = (int)(i >> 12);
    const int c = (int)(i & (DI - 1));
    float acc = conv_b[c];
#pragma unroll
    for (int j = 0; j < 4; ++j) {
        int lj = l - 3 + j;
        if (lj >= 0) acc += xr[(size_t)lj * (2 * DI) + c] * conv_w[j * DI + c];
    }
    xm_out[i] = acc / (1.f + __expf(-acc));  // silu
}

// ======================= pad x_proj_w 160 -> 256 ===========================
__global__ __launch_bounds__(256) void k_pad_xproj(
    const float* __restrict__ w, float* __restrict__ wpad)
{
    const size_t i = (size_t)blockIdx.x * 256 + threadIdx.x;  // over DI*NPAD
    const int k = (int)(i >> 8);
    const int n = (int)(i & (NPAD - 1));
    wpad[i] = (n < DTR + 2 * DS) ? w[(size_t)k * (DTR + 2 * DS) + n] : 0.f;
}

// ======================= softplus(delta_pre + dt_b) ========================
__global__ __launch_bounds__(256) void k_softplus(
    float* __restrict__ delta, const float* __restrict__ dt_b)
{
    const size_t i = (size_t)blockIdx.x * 256 + threadIdx.x;  // over SEQ*DI
    const int c = (int)(i & (DI - 1));
    float v = delta[i] + dt_b[c];
    delta[i] = (v > 20.f) ? v : log1pf(__expf(v));
}

// ======================= selective scan ====================================
// One lane per (channel d, state n): 4096*16 = 65536 lanes = 2048 wave32s.
// Reduction over n (16 lanes) via shfl_xor; lane n==0 stores y.
__global__ __launch_bounds__(256) void k_scan(
    const float* __restrict__ delta,   // 2048 x 4096
    const float* __restrict__ xm,      // 2048 x 4096
    const float* __restrict__ xdbl,    // 2048 x 256 (B at 128..143, C at 144..159)
    const float* __restrict__ A_log,   // 4096 x 16
    float* __restrict__ ys)            // 2048 x (stride 8192), cols 0..4095
{
    const int t = blockIdx.x * 256 + threadIdx.x;
    const int d = t >> 4;
    const int n = t & 15;

    const float Adn = -__expf(A_log[(size_t)d * DS + n]);
    float s = 0.f;

    for (int l = 0; l < SEQ; ++l) {
        const size_t base = (size_t)l * DI + d;
        const float dl  = delta[base];
        const float xv  = xm[base];
        const float Bv  = xdbl[(size_t)l * NPAD + DTR + n];
        const float Cv  = xdbl[(size_t)l * NPAD + DTR + DS + n];
        s = __expf(dl * Adn) * s + dl * Bv * xv;
        float p = s * Cv;
#pragma unroll
        for (int off = 8; off; off >>= 1) p += __shfl_xor(p, off, 16);
        if (n == 0) ys[(size_t)l * (2 * DI) + d] = p;
    }
}

// ======================= gate: y = (ys + xm*D) * silu(r) ===================
__global__ __launch_bounds__(256) void k_gate(
    const float* __restrict__ xr,      // ys in cols [0,4096), r in [4096,8192)
    const float* __restrict__ xm,
    const float* __restrict__ Dw,
    float* __restrict__ yact)          // 2048 x 4096
{
    const size_t i = (size_t)blockIdx.x * 256 + threadIdx.x;  // over SEQ*DI
    const int l = (int)(i >> 12);
    const int c = (int)(i & (DI - 1));
    const float ysv = xr[(size_t)l * (2 * DI) + c];
    const float rv  = xr[(size_t)l * (2 * DI) + DI + c];
    const float y   = ysv + xm[i] * Dw[c];
    yact[i] = y * (rv / (1.f + __expf(-rv)));
}

// ======================= launcher ==========================================
extern "C" void kernel_launch(void* const* d_in, const int* in_sizes, int n_in,
                              void* d_out, int out_size, void* d_ws, size_t ws_size,
                              hipStream_t stream)
{
    const float* x          = (const float*)d_in[0];
    const float* residual   = (const float*)d_in[1];
    const float* in_proj_w  = (const float*)d_in[2];   // 2048 x 8192
    const float* conv_w     = (const float*)d_in[3];   // 4 x 1 x 4096
    const float* conv_b     = (const float*)d_in[4];
    const float* x_proj_w   = (const float*)d_in[5];   // 4096 x 160
    const float* dt_w       = (const float*)d_in[6];   // 128 x 4096
    const float* dt_b       = (const float*)d_in[7];
    const float* A_log      = (const float*)d_in[8];   // 4096 x 16
    const float* Dw         = (const float*)d_in[9];
    const float* out_proj_w = (const float*)d_in[10];  // 4096 x 2048
    const float* norm_scale = (const float*)d_in[11];
    const float* norm_bias  = (const float*)d_in[12];

    float* out = (float*)d_out;                 // 2048 x 2048
    float* res = out + (size_t)SEQ * DM;        // 2048 x 2048
    float* ws  = (float*)d_ws;

    float* h     = ws + WS_H;
    float* xr    = ws + WS_XR;
    float* xm    = ws + WS_XM;
    float* delta = ws + WS_DELTA;   // reused as yact after the scan
    float* xdbl  = ws + WS_XDBL;
    float* xproj = ws + WS_XPROJ;

    // 1) res = x + residual ; h = layernorm(res)
    k_res_ln<<<SEQ, 256, 0, stream>>>(x, residual, norm_scale, norm_bias, res, h);

    // 2) pad x_proj_w to 256 columns
    k_pad_xproj<<<(DI * NPAD) / 256, 256, 0, stream>>>(x_proj_w, xproj);

    // 3) xr = h @ in_proj_w : 2048x2048 * 2048x8192
    k_wmma_gemm_f32<<<dim3(2 * DI / BN, SEQ / BM), 256, 0, stream>>>(
        h, in_proj_w, xr, SEQ, 2 * DI, DM, DM, 2 * DI, 2 * DI);

    // 4) depthwise causal conv + silu on xm half of xr
    k_conv_silu<<<(SEQ * DI) / 256, 256, 0, stream>>>(xr, conv_w, conv_b, xm);

    // 5) x_dbl = xm @ x_proj(padded) : 2048x4096 * 4096x256
    k_wmma_gemm_f32<<<dim3(NPAD / BN, SEQ / BM), 256, 0, stream>>>(
        xm, xproj, xdbl, SEQ, NPAD, DI, DI, NPAD, NPAD);

    // 6) delta_pre = x_dbl[:, :128] @ dt_w : 2048x128 * 128x4096
    k_wmma_gemm_f32<<<dim3(DI / BN, SEQ / BM), 256, 0, stream>>>(
        xdbl, dt_w, delta, SEQ, DI, DTR, NPAD, DI, DI);

    // 7) delta = softplus(delta_pre + dt_b)
    k_softplus<<<(SEQ * DI) / 256, 256, 0, stream>>>(delta, dt_b);

    // 8) selective scan -> ys (stored into xr cols [0,4096), stride 8192)
    k_scan<<<(DI * DS) / 256, 256, 0, stream>>>(delta, xm, xdbl, A_log, xr);

    // 9) yact = (ys + xm*D) * silu(r)  (yact overwrites delta buffer)
    k_gate<<<(SEQ * DI) / 256, 256, 0, stream>>>(xr, xm, Dw, delta);

    // 10) out = yact @ out_proj_w : 2048x4096 * 4096x2048
    k_wmma_gemm_f32<<<dim3(DM / BN, SEQ / BM), 256, 0, stream>>>(
        delta, out_proj_w, out, SEQ, DM, DI, DI, DM, DM);
}